// HyperRadialNeuralFourierCelularAutomata_52604759441998
// MI455X (gfx1250) — compile-verified
//
#include <hip/hip_runtime.h>
#include <math.h>

#define B_     256
#define S_     31
#define P_     961          // S*S
#define HDC_   4096
#define RBF_   5
#define MODES_ 16
#define STEPS_ 8
#define NNZ_   20480        // 5*HDC
#define NELEM_ (RBF_*5*P_)  // 24025 reduction outputs

typedef __attribute__((ext_vector_type(2))) float v2f;
typedef __attribute__((ext_vector_type(8))) float v8f;

// workspace layout (floats)
#define WS_A   0            // 4805: sum_r c3w[r]*mean(probe^2)
#define WS_Bm  4805         // 4805: sum_r c3w[r]*mean(probe)
#define WS_TE  9610         // 256*5 tanh time-encoding gains
#define WS_HS  10890        // 5 hdc row sums (batch 0)

// ---------------------------------------------------------------------------
// Kernel 1: stream rbf_probes (394 MB) once; one wave per (r,d,h,w) element.
// Lanes read consecutive float4 -> 512B fully coalesced per instruction.
// This kernel is the HBM roofline (~17us at 23.3 TB/s); everything else is
// noise.
// ---------------------------------------------------------------------------
__global__ __launch_bounds__(256) void k_probe_reduce(
    const float* __restrict__ probes, const float* __restrict__ c3w,
    float* __restrict__ ws)
{
  int tid  = blockIdx.x * blockDim.x + threadIdx.x;
  int wave = tid >> 5, lane = tid & 31;
  if (wave >= NELEM_) return;
  const float4* p4 = reinterpret_cast<const float4*>(probes + (size_t)wave * HDC_);
  float s1 = 0.f, s2 = 0.f;
  #pragma unroll 4
  for (int k = 0; k < HDC_ / 128; ++k) {           // 32 iters of float4
    float4 v = p4[k * 32 + lane];
    s1 += v.x + v.y + v.z + v.w;
    s2 += v.x * v.x + v.y * v.y + v.z * v.z + v.w * v.w;
  }
  #pragma unroll
  for (int off = 16; off; off >>= 1) {
    s1 += __shfl_xor(s1, off, 32);
    s2 += __shfl_xor(s2, off, 32);
  }
  if (lane == 0) {
    int r = wave / (5 * P_), idx = wave % (5 * P_);
    float c = c3w[r];
    atomicAdd(&ws[WS_A  + idx], c * (s2 * (1.f / HDC_)));
    atomicAdd(&ws[WS_Bm + idx], c * (s1 * (1.f / HDC_)));
  }
}

// ---------------------------------------------------------------------------
// Kernel 2 (tiny): hdc scatter->hsum in LDS, and per-batch time encoding te.
// ---------------------------------------------------------------------------
__global__ __launch_bounds__(256) void k_meta(
    const float* __restrict__ m_in, const float* __restrict__ m_out,
    const int* __restrict__ nzi, const int* __restrict__ sraw,
    const float* __restrict__ ctw, const float* __restrict__ ctb,
    float* __restrict__ ws)
{
  __shared__ signed char sgn[NNZ_];
  __shared__ float hs[5];
  int tid = threadIdx.x;
  for (int i = tid; i < NNZ_; i += 256) sgn[i] = 0;
  if (tid < 5) hs[tid] = 0.f;
  __syncthreads();
  for (int i = tid; i < NNZ_; i += 256) {
    int pos = nzi[i];
    if (pos >= 0 && pos < NNZ_) sgn[pos] = (signed char)(2 * sraw[i] - 1);
  }
  __syncthreads();
  #pragma unroll
  for (int c = 0; c < 5; ++c) {
    float s = 0.f;
    for (int i = tid; i < HDC_; i += 256) s += (float)sgn[c * HDC_ + i];
    atomicAdd(&hs[c], s);
  }
  __syncthreads();
  if (tid < 5) ws[WS_HS + tid] = hs[tid];
  // time encoding: thread tid handles batch tid (B_ == blockDim)
  {
    float tin = m_in[tid] * 0.01f, tout = m_out[tid] * 0.01f;
    float pe[MODES_];
    #pragma unroll
    for (int i = 0; i < MODES_; ++i) {
      float div  = powf(10000.f, (2.f * (float)(i >> 1)) / (float)MODES_);
      float ain  = tin  / div;
      float aout = tout / div;
      pe[i] = ((i & 1) == 0) ? 0.5f * (sinf(ain) + cosf(ain))
                             : 0.5f * (cosf(ain) + sinf(aout));
    }
    #pragma unroll
    for (int r = 0; r < 5; ++r) {
      float a = ctb[r];
      #pragma unroll
      for (int m = 0; m < MODES_; ++m) a += pe[m] * ctw[r * MODES_ + m];
      ws[WS_TE + tid * 5 + r] = tanhf(a);
    }
  }
}

// ---------------------------------------------------------------------------
// Kernel 3: per-batch fused RBF-compress prologue + 8-step NCA via WMMA
// f32 16x16x4 implicit GEMM + 1x1 output projection.
// h kept double-buffered in LDS with a zero halo and a 6th zero channel
// plane so the im2col gather is 24 unconditional ds_load_b32 per tile.
// ---------------------------------------------------------------------------
#define CH_   (33 * 34)     // one channel plane (with halo, W padded to 34)
#define BUF_  (6 * CH_)     // 5 channels + 1 zero plane

__global__ __launch_bounds__(256) void k_nca(
    const float* __restrict__ din, const float* __restrict__ sin_,
    const float* __restrict__ spk,
    const float* __restrict__ c3w, const float* __restrict__ c3b,
    const float* __restrict__ ncaw, const float* __restrict__ ncab,
    const float* __restrict__ outw, const float* __restrict__ outb,
    const float* __restrict__ ws, float* __restrict__ dout)
{
  __shared__ float hb[2 * BUF_];       // ~52.6 KB
  int b    = blockIdx.x;
  int tid  = threadIdx.x;
  int lane = tid & 31, wave = tid >> 5;

  // zero everything once: halos and channel-5 zero plane stay 0 forever
  for (int i = tid; i < 2 * BUF_; i += 256) hb[i] = 0.f;

  float te5[5], H5[5];
  float Cw  = c3w[0] + c3w[1] + c3w[2] + c3w[3] + c3w[4];
  float cb3 = c3b[0];
  #pragma unroll
  for (int d = 0; d < 5; ++d) {
    te5[d] = ws[WS_TE + b * 5 + d];
    H5[d]  = (b == 0) ? ws[WS_HS + d] : 0.f;
  }
  __syncthreads();

  // prologue: x = tanh(A - 2*proj*Bm + Cw*proj^2 + c3b) into buffer 0 interior
  for (int p = tid; p < P_; p += 256) {
    int y = p / S_, x = p - y * S_;
    float raw[5];
    #pragma unroll
    for (int c = 0; c < 4; ++c)
      raw[c] = din[(size_t)b * 4 * P_ + (size_t)(c * S_ + y) * S_ + x];
    raw[4] = sin_[(size_t)b * P_ + p];
    float dsum = 0.f;
    #pragma unroll
    for (int c = 0; c < 5; ++c) dsum += raw[c] * (1.f + te5[c]);
    #pragma unroll
    for (int d = 0; d < 5; ++d) {
      float proj = H5[d] * dsum;
      float val  = ws[WS_A + d * P_ + p] - 2.f * proj * ws[WS_Bm + d * P_ + p]
                 + Cw * proj * proj + cb3;
      hb[d * CH_ + (y + 1) * 34 + (x + 1)] = tanhf(val);
    }
  }

  // A-fragments (weights): 16x4 f32 layout -> lanes 0-15: K=4j+{0,1},
  // lanes 16-31: K=4j+{2,3}; rows M>=5 and taps k>=45 are zero.
  int M = lane & 15;
  int khalf = (lane >> 4) << 1;
  v2f af[12];
  int koff[24];                         // per-lane LDS float offsets per tap
  #pragma unroll
  for (int j = 0; j < 12; ++j) {
    #pragma unroll
    for (int t = 0; t < 2; ++t) {
      int k = j * 4 + khalf + t;        // 0..47
      af[j][t] = (M < 5 && k < 45) ? ncaw[M * 45 + k] : 0.f;
      int c = k / 9, r9 = k - c * 9;    // k>=45 -> c==5 (zero plane)
      int ky = r9 / 3, kx = r9 - ky * 3;
      koff[j * 2 + t] = c * CH_ + ky * 34 + kx;
    }
  }
  float nb5[5];
  #pragma unroll
  for (int o = 0; o < 5; ++o) nb5[o] = ncab[o];
  __syncthreads();

  int cur = 0;
  for (int step = 0; step < STEPS_; ++step) {
    float pg  = spk[step];
    int   nxt = cur ^ 1;
    const float* src = hb + cur * BUF_;
    float*       dst = hb + nxt * BUF_;
    for (int tile = wave; tile < (P_ + 15) / 16; tile += 8) {
      int pix   = tile * 16 + M;
      bool live = pix < P_;
      int pixc  = live ? pix : P_ - 1;  // clamp: garbage cols never stored
      int py = pixc / S_, px = pixc - py * S_;
      int base = py * 34 + px;          // reads cover rows py..py+2 (halo)

      float bv[24];                     // branch-free batched gather
      #pragma unroll
      for (int i = 0; i < 24; ++i) bv[i] = src[base + koff[i]];

      v8f acc = {};
      #pragma unroll
      for (int j = 0; j < 12; ++j) {
        v2f bf; bf[0] = bv[2 * j]; bf[1] = bv[2 * j + 1];
        acc = __builtin_amdgcn_wmma_f32_16x16x4_f32(
                  false, af[j], false, bf, (short)0, acc, false, false);
      }
      // D layout: lane<16 holds D[M=o][N=lane] in acc[o]
      if (lane < 16 && live) {
        int wb = (py + 1) * 34 + (px + 1);
        #pragma unroll
        for (int o = 0; o < 5; ++o) {
          float h_old = src[o * CH_ + wb];
          dst[o * CH_ + wb] = h_old + tanhf(acc[o] + nb5[o]) * pg;
        }
      }
    }
    __syncthreads();
    cur = nxt;
  }

  // epilogue: 1x1 conv 5->5, write 5 concatenated (B,S,S) outputs
  const float* fin = hb + cur * BUF_;
  for (int p = tid; p < P_; p += 256) {
    int y = p / S_, x = p - y * S_;
    float cc[5];
    #pragma unroll
    for (int c = 0; c < 5; ++c) cc[c] = fin[c * CH_ + (y + 1) * 34 + (x + 1)];
    #pragma unroll
    for (int o = 0; o < 5; ++o) {
      float v = outb[o];
      #pragma unroll
      for (int c = 0; c < 5; ++c) v += outw[o * 5 + c] * cc[c];
      dout[(size_t)o * B_ * P_ + (size_t)b * P_ + p] = v;
    }
  }
}

extern "C" void kernel_launch(void* const* d_in, const int* in_sizes, int n_in,
                              void* d_out, int out_size, void* d_ws, size_t ws_size,
                              hipStream_t stream) {
  const float* din    = (const float*)d_in[0];
  const float* sin_   = (const float*)d_in[1];
  const float* m_in   = (const float*)d_in[2];
  const float* m_out  = (const float*)d_in[3];
  const float* spk    = (const float*)d_in[4];
  const int*   nzi    = (const int*)  d_in[5];
  const int*   sraw   = (const int*)  d_in[6];
  const float* probes = (const float*)d_in[7];
  const float* ctw    = (const float*)d_in[8];
  const float* ctb    = (const float*)d_in[9];
  const float* c3w    = (const float*)d_in[10];
  const float* c3b    = (const float*)d_in[11];
  const float* ncaw   = (const float*)d_in[12];
  const float* ncab   = (const float*)d_in[13];
  const float* outw   = (const float*)d_in[14];
  const float* outb   = (const float*)d_in[15];
  float* ws  = (float*)d_ws;
  float* out = (float*)d_out;

  // zero the atomic accumulators (A and Bm) — capture-safe memset node
  hipMemsetAsync(ws, 0, (size_t)(2 * 5 * P_) * sizeof(float), stream);

  int blocks1 = (NELEM_ * 32 + 255) / 256;   // one wave32 per reduction element
  k_probe_reduce<<<blocks1, 256, 0, stream>>>(probes, c3w, ws);
  k_meta<<<1, 256, 0, stream>>>(m_in, m_out, nzi, sraw, ctw, ctb, ws);
  k_nca<<<B_, 256, 0, stream>>>(din, sin_, spk, c3w, c3b, ncaw, ncab,
                                outw, outb, ws, out);
}